// PinnGenerator_13408887898247
// MI455X (gfx1250) — compile-verified
//
#include <hip/hip_runtime.h>
#include <stdint.h>

// ---------------------------------------------------------------------------
// 21x Jacobi relaxation (5-point, zero Dirichlet) on 16 x 1022 x 1022 fp32.
// Per-wave 16x16 tiles; 18x18 halo slab staged in LDS via CDNA5 async-to-LDS;
// neighbor sum computed as M*Y + Y*M via chained V_WMMA_F32_16X16X4_F32.
// ---------------------------------------------------------------------------

#define N_INT   1022      // interior grid (1022 x 1022)
#define NIMG    16
#define TILE    16
#define TPI     64        // tiles per image dim (64*16 = 1024 covers 1022)
#define WPB     8         // waves per block (256 threads)
#define LDSW    20        // LDS row stride in floats (18 used + pad)
#define SLAB    384       // floats per wave slab (12 * 32, full-EXEC fill)

typedef __attribute__((ext_vector_type(2))) float v2f;
typedef __attribute__((ext_vector_type(8))) float v8f;
typedef __attribute__((address_space(1))) int gds_i32;   // global int
typedef __attribute__((address_space(3))) int lds_i32;   // LDS int

#ifndef __has_builtin
#define __has_builtin(x) 0
#endif

#if __has_builtin(__builtin_amdgcn_wmma_f32_16x16x4_f32)
#define HAVE_WMMA_F32 1
#else
#define HAVE_WMMA_F32 0
#endif

#if __has_builtin(__builtin_amdgcn_global_load_async_to_lds_b32)
#define HAVE_ASYNC_BI 1
#else
#define HAVE_ASYNC_BI 0
#endif

__device__ __forceinline__ void async_ld_f32(const float* gsrc, float* lds_dst) {
#if HAVE_ASYNC_BI
  __builtin_amdgcn_global_load_async_to_lds_b32(
      (gds_i32*)(uintptr_t)gsrc,
      (lds_i32*)(uint32_t)(uintptr_t)lds_dst,
      0, 0);
#else
  uint32_t l = (uint32_t)(uintptr_t)lds_dst;
  asm volatile("global_load_async_to_lds_b32 %0, %1, off"
               :: "v"(l), "v"(gsrc) : "memory");
#endif
}

__device__ __forceinline__ void wait_async0() {
#if __has_builtin(__builtin_amdgcn_s_wait_asynccnt)
  __builtin_amdgcn_s_wait_asynccnt(0);
#else
  asm volatile("s_wait_asynccnt 0x0" ::: "memory");
#endif
  asm volatile("" ::: "memory");
}

#define YLDS(rr, cc) slab[((rr) + 1) * LDSW + ((cc) + 1)]

__global__ __launch_bounds__(256) void jac_step_kernel(
    const float* __restrict__ xin,  // NIMG x 1022 x 1022 (current iterate)
    const float* __restrict__ f,    // NIMG x 1024 x 1024
    const float* __restrict__ mu,   // 1
    const float* __restrict__ zp,   // zeroed scratch (safe OOB source)
    float* __restrict__ yout)       // NIMG x 1022 x 1022
{
  __shared__ float slab_all[WPB * SLAB];
  const int lane = threadIdx.x & 31;
  const int wave = threadIdx.x >> 5;
  float* slab = &slab_all[wave * SLAB];

  const int t   = blockIdx.x * WPB + wave;  // grid sized exactly: t < 65536
  const int img = t >> 12;                  // / (TPI*TPI)
  const int rem = t & 4095;
  const int r0  = (rem >> 6) * TILE;
  const int c0  = (rem & 63) * TILE;

  const float* X = xin + img * (N_INT * N_INT);

  // ---- async fill of 18x18 halo tile (zero page for out-of-domain) --------
  // 12 iterations, all 32 lanes always active (full EXEC).
  #pragma unroll
  for (int e = lane; e < SLAB; e += 32) {
    const int i = e / LDSW;
    const int j = e - i * LDSW;
    const int r = r0 - 1 + i;
    const int c = c0 - 1 + j;
    const bool ok = (i < 18) & (j < 18) &
                    (r >= 0) & (r < N_INT) & (c >= 0) & (c < N_INT);
    const float* src = ok ? (X + r * N_INT + c) : zp;
    async_ld_f32(src, &slab[e]);
  }
  wait_async0();

  // ---- constant-term accumulator C = 0.25*(f*H^2/mu [+ halo]) -------------
  const float muv    = mu[0];
  const float HH     = (float)(1.0 / (1023.0 * 1023.0));
  const float fscale = HH / (4.0f * muv);

  const int hrow = (lane < 16) ? 0 : 8;  // C/D layout: vgpr r -> rows r, r+8
  const int col  = lane & 15;
  const float* F = f + img * 1024 * 1024;

  v8f acc;
  #pragma unroll
  for (int rr = 0; rr < 8; ++rr) {
    const int row = rr + hrow;
    int fr = r0 + row + 1; if (fr > 1023) fr = 1023;   // clamp: OOB rows never stored
    int fc = c0 + col + 1; if (fc > 1023) fc = 1023;
    float cval = F[fr * 1024 + fc] * fscale;
#if HAVE_WMMA_F32
    float h = 0.0f;                    // halo terms the in-tile matmul misses
    if (row == 0)  h += YLDS(-1, col);
    if (row == 15) h += YLDS(16, col);
    if (col == 0)  h += YLDS(row, -1);
    if (col == 15) h += YLDS(row, 16);
    cval += 0.25f * h;
#endif
    acc[rr] = cval;
  }

#if HAVE_WMMA_F32
  // ---- D = M*Y + Y*M + C ; M = 0.25 * tridiagonal(0-diag), exact in fp32 --
  const int arow = lane & 15;               // A layout: lanes hold matrix rows
  const int koff = (lane < 16) ? 0 : 2;     // K slots 0,1 | 2,3 per half-wave
  v8f acc2 = {0.f, 0.f, 0.f, 0.f, 0.f, 0.f, 0.f, 0.f};
  #pragma unroll
  for (int q = 0; q < 4; ++q) {
    const int kb = 4 * q + koff;
    // vertical sum: A = M chunk (16x4), B = Y row-chunk (4x16)
    v2f aM;
    { int d0 = arow - kb;       if (d0 < 0) d0 = -d0;
      int d1 = arow - (kb + 1); if (d1 < 0) d1 = -d1;
      aM.x = (d0 == 1) ? 0.25f : 0.0f;
      aM.y = (d1 == 1) ? 0.25f : 0.0f; }
    v2f bY; bY.x = YLDS(kb, col); bY.y = YLDS(kb + 1, col);
    acc = __builtin_amdgcn_wmma_f32_16x16x4_f32(
        false, aM, false, bY, (short)0, acc, false, false);

    // horizontal sum: A = Y col-chunk (16x4), B = M chunk (4x16)
    v2f aY; aY.x = YLDS(arow, kb); aY.y = YLDS(arow, kb + 1);
    v2f bM;
    { int d0 = kb - col;       if (d0 < 0) d0 = -d0;
      int d1 = (kb + 1) - col; if (d1 < 0) d1 = -d1;
      bM.x = (d0 == 1) ? 0.25f : 0.0f;
      bM.y = (d1 == 1) ? 0.25f : 0.0f; }
    acc2 = __builtin_amdgcn_wmma_f32_16x16x4_f32(
        false, aY, false, bM, (short)0, acc2, false, false);
  }
  #pragma unroll
  for (int rr = 0; rr < 8; ++rr) acc[rr] += acc2[rr];
#else
  // ---- VALU fallback: direct 5-point stencil from the LDS slab ------------
  #pragma unroll
  for (int rr = 0; rr < 8; ++rr) {
    const int row = rr + hrow;
    acc[rr] += 0.25f * (YLDS(row - 1, col) + YLDS(row + 1, col) +
                        YLDS(row, col - 1) + YLDS(row, col + 1));
  }
#endif

  // ---- predicated store (skip the 2-row/col overhang of the 1024 tiling) --
  float* O = yout + img * (N_INT * N_INT);
  #pragma unroll
  for (int rr = 0; rr < 8; ++rr) {
    const int row = rr + hrow;
    const int gr = r0 + row, gc = c0 + col;
    if (gr < N_INT && gc < N_INT) O[gr * N_INT + gc] = acc[rr];
  }
}

__global__ void zero_pad_kernel(float* zpk) { zpk[threadIdx.x] = 0.0f; }

extern "C" void kernel_launch(void* const* d_in, const int* in_sizes, int n_in,
                              void* d_out, int out_size, void* d_ws, size_t ws_size,
                              hipStream_t stream) {
  (void)in_sizes; (void)n_in; (void)out_size; (void)ws_size;
  const float* pre = (const float*)d_in[1];  // d_in[0] = x is unused by reference
  const float* f   = (const float*)d_in[2];
  const float* mu  = (const float*)d_in[3];
  float* out = (float*)d_out;
  float* zp  = (float*)d_ws;                        // 1 KB zero page
  float* W   = (float*)((char*)d_ws + 4096);        // ping-pong buffer (~64 MB)

  zero_pad_kernel<<<1, 256, 0, stream>>>(zp);

  const int tiles  = NIMG * TPI * TPI;   // 65536
  const int blocks = tiles / WPB;        // 8192

  // 21 steps total: even steps write d_out, odd write W; step 20 lands in d_out.
  const float* src = pre;
  for (int s = 0; s < 21; ++s) {
    float* dst = ((s & 1) == 0) ? out : W;
    jac_step_kernel<<<blocks, 256, 0, stream>>>(src, f, mu, zp, dst);
    src = dst;
  }
}